// BBoxProcessor_21887153340661
// MI455X (gfx1250) — compile-verified
//
#include <hip/hip_runtime.h>
#include <stdint.h>

// ---------------------------------------------------------------------------
// BBoxProcessor: stable stream compaction of (1, R, 6) detections.
//   valid      = (class_id == 0) && (score >= 0.35)
//   valid_full = (score >= 0.35)
// Outputs (flat, in order): rois (R*5), rois_full (R*6), n (1), n_full (1).
// Memory-bound (0 FLOPs, ~0.5 GB traffic -> ~20us @ 23.3 TB/s): wave32 ballot
// scans + CDNA5 async global->LDS staging (global_load_async_to_lds_*,
// s_wait_asynccnt) with a 2-deep software pipeline. Input (96 MB) stays
// resident in the 192 MB L2 between the count and scatter passes.
// ---------------------------------------------------------------------------

#define SCORE_THRESH 0.35f
#define BLOCK 256
#define ITEMS 16
#define CHUNK 256                        // rows per pipelined chunk (== BLOCK)
#define ROWS_PER_BLOCK (BLOCK * ITEMS)   // 4096
#define NWAVES (BLOCK / 32)

#ifndef USE_ASYNC_LDS
#define USE_ASYNC_LDS 1
#endif

__device__ __forceinline__ unsigned ballot32(bool p) {
#if __has_builtin(__builtin_amdgcn_ballot_w32)
    return __builtin_amdgcn_ballot_w32(p);
#else
    return (unsigned)__ballot(p);
#endif
}

// -------------------------------- pass 3 -----------------------------------
// (Placed first in the file so the disasm snippet shows the async-LDS path.)
__global__ __launch_bounds__(BLOCK) void bbox_scatter_kernel(
    const float* __restrict__ det, int R,
    const int* __restrict__ off_v, const int* __restrict__ off_f,
    float* __restrict__ rois, float* __restrict__ rois_full) {
#if USE_ASYNC_LDS
    __shared__ float buf[2][CHUNK * 6];   // 12 KB double buffer, 24 B/row
#endif
    __shared__ int wv[NWAVES], wf[NWAVES];
    const int tid  = threadIdx.x;
    const int lane = tid & 31;
    const int wid  = tid >> 5;
    const unsigned lt_mask = (1u << lane) - 1u;
    const int blockBase = blockIdx.x * ROWS_PER_BLOCK;

    int base_v = off_v[blockIdx.x];
    int base_f = off_f[blockIdx.x];

#if USE_ASYNC_LDS
    // prime the pipeline: chunk 0 -> buf[0] (each lane async-copies its own
    // 24 B row: one b128 + one b64; ASYNCcnt tracks completion in order)
    {
        int row = blockBase + tid;
        if (row < R) {
            unsigned lds = (unsigned)(uintptr_t)&buf[0][tid * 6];
            const char* g = (const char*)det + (size_t)row * 24;
            asm volatile("global_load_async_to_lds_b128 %0, %1, off"
                         :: "v"(lds), "v"(g) : "memory");
            asm volatile("global_load_async_to_lds_b64 %0, %1, off offset:16"
                         :: "v"(lds), "v"(g) : "memory");
        }
    }
#endif

    for (int k = 0; k < ITEMS; ++k) {
        const int sel = k & 1;
#if USE_ASYNC_LDS
        if (k + 1 < ITEMS) {  // prefetch chunk k+1 into the other buffer
            int nrow = blockBase + (k + 1) * CHUNK + tid;
            if (nrow < R) {
                unsigned lds = (unsigned)(uintptr_t)&buf[sel ^ 1][tid * 6];
                const char* g = (const char*)det + (size_t)nrow * 24;
                asm volatile("global_load_async_to_lds_b128 %0, %1, off"
                             :: "v"(lds), "v"(g) : "memory");
                asm volatile("global_load_async_to_lds_b64 %0, %1, off offset:16"
                             :: "v"(lds), "v"(g) : "memory");
            }
            asm volatile("s_wait_asynccnt 2" ::: "memory");  // chunk k complete
        } else {
            asm volatile("s_wait_asynccnt 0" ::: "memory");
        }
#endif
        __syncthreads();

        int row = blockBase + k * CHUNK + tid;
        bool inr = row < R;
        float r0 = 0.f, r1 = 0.f, r2 = 0.f, r3 = 0.f, r4 = 0.f, r5 = 0.f;
        if (inr) {
#if USE_ASYNC_LDS
            const float* p = &buf[sel][tid * 6];
#else
            const float* p = det + (size_t)row * 6;
#endif
            r0 = p[0]; r1 = p[1]; r2 = p[2]; r3 = p[3]; r4 = p[4]; r5 = p[5];
        }
        bool pf = inr && (r5 >= SCORE_THRESH);
        bool pv = pf && (r0 == 0.0f);

        unsigned mv = ballot32(pv);
        unsigned mf = ballot32(pf);
        if (lane == 0) { wv[wid] = __popc(mv); wf[wid] = __popc(mf); }
        __syncthreads();

        int pre_v = 0, pre_f = 0, tot_v = 0, tot_f = 0;
        for (int w = 0; w < NWAVES; ++w) {
            int a = wv[w], b = wf[w];
            if (w < wid) { pre_v += a; pre_f += b; }
            tot_v += a; tot_f += b;
        }

        if (pv) {
            int dst = base_v + pre_v + __popc(mv & lt_mask);
            float* o = rois + (size_t)dst * 5;   // o[0] stays 0 from memset
            o[1] = r1; o[2] = r2; o[3] = r3; o[4] = r4;
        }
        if (pf) {
            int dst = base_f + pre_f + __popc(mf & lt_mask);
            float* o = rois_full + (size_t)dst * 6;
            o[0] = r0; o[1] = r1; o[2] = r2; o[3] = r3; o[4] = r4; o[5] = r5;
        }
        base_v += tot_v;
        base_f += tot_f;
        __syncthreads();  // wv/wf + LDS buffer reuse protection
    }
}

// -------------------------------- pass 1 -----------------------------------
__global__ __launch_bounds__(BLOCK) void bbox_count_kernel(
    const float* __restrict__ det, int R,
    int* __restrict__ cnt_v, int* __restrict__ cnt_f) {
    __shared__ int swv[NWAVES], swf[NWAVES];
    const int tid  = threadIdx.x;
    const int lane = tid & 31;
    const int wid  = tid >> 5;
    const int base = blockIdx.x * ROWS_PER_BLOCK;

    int cv = 0, cf = 0;
    for (int k = 0; k < ITEMS; ++k) {
        int row = base + k * CHUNK + tid;
        if (row < R) {
            const float* p = det + (size_t)row * 6;
            if (k + 2 < ITEMS)  // pull future cachelines into GL2 (global_prefetch_b8)
                __builtin_prefetch(p + 2 * CHUNK * 6, 0, 1);
            float cls = p[0];
            float sc  = p[5];
            bool pf = (sc >= SCORE_THRESH);
            bool pv = pf && (cls == 0.0f);
            cf += pf ? 1 : 0;
            cv += pv ? 1 : 0;
        }
    }
    // wave32 reduction
    for (int d = 16; d > 0; d >>= 1) {
        cv += __shfl_down(cv, d, 32);
        cf += __shfl_down(cf, d, 32);
    }
    if (lane == 0) { swv[wid] = cv; swf[wid] = cf; }
    __syncthreads();
    if (tid == 0) {
        int tv = 0, tf = 0;
        for (int w = 0; w < NWAVES; ++w) { tv += swv[w]; tf += swf[w]; }
        cnt_v[blockIdx.x] = tv;
        cnt_f[blockIdx.x] = tf;
    }
}

// -------------------------------- pass 2 -----------------------------------
// Single 1024-thread block scans <=1024 block counts (R <= 1024*4096).
__global__ __launch_bounds__(1024) void bbox_scan_kernel(
    const int* __restrict__ cnt_v, const int* __restrict__ cnt_f,
    int* __restrict__ off_v, int* __restrict__ off_f,
    int nblk, float* __restrict__ out_counts) {
    __shared__ int wsv[32], wsf[32];
    const int tid  = threadIdx.x;
    const int lane = tid & 31;
    const int wid  = tid >> 5;

    int cv = (tid < nblk) ? cnt_v[tid] : 0;
    int cf = (tid < nblk) ? cnt_f[tid] : 0;
    int iv = cv, jf = cf;
    for (int d = 1; d < 32; d <<= 1) {  // wave32 inclusive scan
        int tv = __shfl_up(iv, d, 32);
        int tf = __shfl_up(jf, d, 32);
        if (lane >= d) { iv += tv; jf += tf; }
    }
    if (lane == 31) { wsv[wid] = iv; wsf[wid] = jf; }
    __syncthreads();
    if (wid == 0) {  // scan the 32 wave totals
        int a = wsv[lane], b = wsf[lane];
        for (int d = 1; d < 32; d <<= 1) {
            int ta = __shfl_up(a, d, 32);
            int tb = __shfl_up(b, d, 32);
            if (lane >= d) { a += ta; b += tb; }
        }
        wsv[lane] = a; wsf[lane] = b;
    }
    __syncthreads();
    if (wid > 0) { iv += wsv[wid - 1]; jf += wsf[wid - 1]; }
    if (tid < nblk) {
        off_v[tid] = iv - cv;   // exclusive
        off_f[tid] = jf - cf;
    }
    if (tid == 1023) {          // grand totals (padding lanes contribute 0)
        out_counts[0] = (float)iv;
        out_counts[1] = (float)jf;
    }
}

// ---------------------------------------------------------------------------
extern "C" void kernel_launch(void* const* d_in, const int* in_sizes, int n_in,
                              void* d_out, int out_size, void* d_ws, size_t ws_size,
                              hipStream_t stream) {
    (void)n_in; (void)ws_size;
    const float* det = (const float*)d_in[0];
    const int R    = in_sizes[0] / 6;
    const int nblk = (R + ROWS_PER_BLOCK - 1) / ROWS_PER_BLOCK;  // 1024 for R=4M

    int* cnt_v = (int*)d_ws;
    int* cnt_f = cnt_v + nblk;
    int* off_v = cnt_f + nblk;
    int* off_f = off_v + nblk;

    float* out       = (float*)d_out;
    float* rois      = out;                     // R*5
    float* rois_full = out + (size_t)R * 5;     // R*6
    float* counts    = out + (size_t)R * 11;    // n, n_full

    // Zero entire output: gives rois col-0 zeros and the compacted tails.
    hipMemsetAsync(d_out, 0, (size_t)out_size * sizeof(float), stream);

    bbox_count_kernel<<<nblk, BLOCK, 0, stream>>>(det, R, cnt_v, cnt_f);
    bbox_scan_kernel<<<1, 1024, 0, stream>>>(cnt_v, cnt_f, off_v, off_f, nblk, counts);
    bbox_scatter_kernel<<<nblk, BLOCK, 0, stream>>>(det, R, off_v, off_f, rois, rois_full);
}